// FlowLoss_58102317580772
// MI455X (gfx1250) — compile-verified
//
#include <hip/hip_runtime.h>

#define NUM_NODES 100000

typedef __attribute__((ext_vector_type(2))) float v2f;
typedef __attribute__((ext_vector_type(4))) float v4f;
typedef __attribute__((ext_vector_type(8))) float v8f;
typedef __attribute__((ext_vector_type(4))) int   v4i;

// ---------------------------------------------------------------------------
// Kernel 1: zero the node accumulator (in d_ws) and the scalar output.
// ---------------------------------------------------------------------------
__global__ __launch_bounds__(256) void flow_zero_kernel(float* __restrict__ node,
                                                        float* __restrict__ out,
                                                        int n) {
    int i = blockIdx.x * blockDim.x + threadIdx.x;
    if (i < n) node[i] = 0.0f;
    if (i == 0) out[0] = 0.0f;
}

// ---------------------------------------------------------------------------
// Kernel 2: fused signed scatter. node[dst] += v ; node[src] -= v.
// Streams edge_index (51.2 MB) + y_hat (25.6 MB) with non-temporal 128-bit
// loads; atomics hit a 400 KB accumulator that stays resident in the 192 MB
// L2, so HBM sees only the one-shot stream (~77 MB -> ~3.3 us at 23.3 TB/s).
// ---------------------------------------------------------------------------
__global__ __launch_bounds__(256) void flow_scatter_kernel(const int*   __restrict__ src,
                                                           const int*   __restrict__ dst,
                                                           const float* __restrict__ val,
                                                           float*       __restrict__ node,
                                                           int E) {
    int t    = blockIdx.x * blockDim.x + threadIdx.x;
    int base = t * 4;
    if (base + 3 < E) {
        v4i s = __builtin_nontemporal_load((const v4i*)(src + base));
        v4i d = __builtin_nontemporal_load((const v4i*)(dst + base));
        v4f v = __builtin_nontemporal_load((const v4f*)(val + base));
        atomicAdd(node + d[0],  v[0]);
        atomicAdd(node + d[1],  v[1]);
        atomicAdd(node + d[2],  v[2]);
        atomicAdd(node + d[3],  v[3]);
        atomicAdd(node + s[0], -v[0]);
        atomicAdd(node + s[1], -v[1]);
        atomicAdd(node + s[2], -v[2]);
        atomicAdd(node + s[3], -v[3]);
    } else {
        for (int i = base; i < E; ++i) {
            float v = val[i];
            atomicAdd(node + dst[i],  v);
            atomicAdd(node + src[i], -v);
        }
    }
}

// ---------------------------------------------------------------------------
// Kernel 3: sum(|node|)/N via V_WMMA_F32_16X16X4_F32.
// Full chunks (64 elems) are loaded branch-free as one float2 per lane
// (coalesced global_load_b64) -> A tile (16x4 f32, 2 VGPRs/lane). B is all
// ones, so D = A*1 + C accumulates row-sums into C; a total sum is
// permutation-invariant so the element->slot mapping is irrelevant. Every
// column of C is identical => full 32-lane shuffle sum = 16 * chunk_total.
// The <64-elem tail is added by wave 0 pre-scaled by 16 so a single /16
// at the end is exact for both parts. No divergent control flow touches the
// WMMA loop, so EXEC stays all-ones as the ISA requires.
// ---------------------------------------------------------------------------
__global__ __launch_bounds__(256) void flow_reduce_wmma_kernel(const float* __restrict__ node,
                                                               float*       __restrict__ out,
                                                               int n) {
    const int lane          = threadIdx.x & 31;
    const int waveInBlock   = threadIdx.x >> 5;
    const int wavesPerBlock = blockDim.x >> 5;
    const int gwave         = blockIdx.x * wavesPerBlock + waveInBlock;
    const int totalWaves    = gridDim.x * wavesPerBlock;
    const int numFull       = n >> 6;          // full 64-element chunks

    v8f c = {};                 // 16x16 f32 accumulator, starts at zero
    v2f ones;
    ones[0] = 1.0f;
    ones[1] = 1.0f;

    for (int ch = gwave; ch < numFull; ch += totalWaves) {
        // lane reads node[ch*64 + 2*lane .. +1] : contiguous 256B per wave
        v2f a = *((const v2f*)(node + (ch << 6)) + lane);
        a[0] = __builtin_fabsf(a[0]);
        a[1] = __builtin_fabsf(a[1]);
        // 8 args: (neg_a, A, neg_b, B, c_mod, C, reuse_a, reuse_b)
        c = __builtin_amdgcn_wmma_f32_16x16x4_f32(
            /*neg_a=*/false, a, /*neg_b=*/false, ones,
            /*c_mod=*/(short)0, c, /*reuse_a=*/false, /*reuse_b=*/false);
    }

    float s = c[0] + c[1] + c[2] + c[3] + c[4] + c[5] + c[6] + c[7];

    // tail (n % 64 elements), handled by wave 0 only; pre-scale by 16 so the
    // final /16 (which removes the WMMA column duplication) is exact here too.
    if (gwave == 0) {
        int idx = (numFull << 6) + lane;
        if (idx < n)      s += 16.0f * __builtin_fabsf(node[idx]);
        idx += 32;
        if (idx < n)      s += 16.0f * __builtin_fabsf(node[idx]);
    }

    // full-wave reduction (wave32)
    for (int off = 16; off > 0; off >>= 1)
        s += __shfl_down(s, off, 32);

    if (lane == 0) {
        atomicAdd(out, s * (1.0f / 16.0f) * (1.0f / (float)n));
    }
}

// ---------------------------------------------------------------------------
// Launch
// ---------------------------------------------------------------------------
extern "C" void kernel_launch(void* const* d_in, const int* in_sizes, int n_in,
                              void* d_out, int out_size, void* d_ws, size_t ws_size,
                              hipStream_t stream) {
    const int*   edge = (const int*)d_in[0];     // [2, E] flattened: row0=src, row1=dst
    const float* yhat = (const float*)d_in[1];   // [E, 1]
    const int E = in_sizes[1];                   // y_hat element count == E

    const int*   src  = edge;
    const int*   dst  = edge + E;
    float*       node = (float*)d_ws;            // NUM_NODES floats of scratch
    float*       out  = (float*)d_out;

    // 1) zero accumulator + output
    {
        int threads = 256;
        int blocks  = (NUM_NODES + threads - 1) / threads;
        flow_zero_kernel<<<blocks, threads, 0, stream>>>(node, out, NUM_NODES);
    }

    // 2) signed scatter (4 edges per thread)
    {
        int threads  = 256;
        int nThreads = (E + 3) / 4;
        int blocks   = (nThreads + threads - 1) / threads;
        flow_scatter_kernel<<<blocks, threads, 0, stream>>>(src, dst, yhat, node, E);
    }

    // 3) WMMA abs-sum reduction -> mean
    {
        int threads = 256;   // 8 waves/block
        int blocks  = 32;    // 256 waves total, ~6 full chunks each
        flow_reduce_wmma_kernel<<<blocks, threads, 0, stream>>>(node, out, NUM_NODES);
    }
}